// DirectSphereProjection_9285719294013
// MI455X (gfx1250) — compile-verified
//
#include <hip/hip_runtime.h>

typedef __attribute__((ext_vector_type(2))) float v2f;
typedef __attribute__((ext_vector_type(8))) float v8f;

#define IMG_H   192
#define IMG_W   192
#define HWPIX   (IMG_H * IMG_W)
#define HIDDEN  16
#define NPOSE   4

// Raw fire-and-forget f32 atomic add; bypasses the compiler's atomic-optimizer
// lane-scan expansion (we already do our own wave-level reduction).
__device__ __forceinline__ void atomic_add_f32(float* p, float v) {
    asm volatile("global_atomic_add_f32 %0, %1, off scope:SCOPE_DEV"
                 :: "v"(p), "v"(v)
                 : "memory");
}

// Single VOP3 instruction: e = x + |x|  (== 2x if x>0 else 0, i.e. 2*relu(x)).
// Inline asm pins the |src| modifier; the pk-vectorizer would otherwise expand
// the abs into explicit v_and_b32 masks.
__device__ __forceinline__ float add_abs(float x) {
    float e;
    asm("v_add_f32 %0, %1, |%1|" : "=v"(e) : "v"(x));
    return e;
}

// Cross-half add without touching the LDS pipe: v_permlanex16_b32 with
// identity lane selects makes every lane read lane (l ^ 16) of the other row.
__device__ __forceinline__ float xor16_add(float x) {
    int t = __builtin_amdgcn_permlanex16(
        0, __builtin_bit_cast(int, x),
        0x76543210, (int)0xfedcba98u, false, false);
    return x + __builtin_bit_cast(float, t);
}

// Sum x across each row of 16 lanes using pure-VALU DPP ops.
// After this every lane holds the sum of its 16-lane row.
__device__ __forceinline__ float row16_reduce_add(float x) {
    int t;
    // quad_perm [1,0,3,2] : xor 1
    t = __builtin_amdgcn_update_dpp(0, __builtin_bit_cast(int, x), 0xB1, 0xF, 0xF, true);
    x += __builtin_bit_cast(float, t);
    // quad_perm [2,3,0,1] : xor 2
    t = __builtin_amdgcn_update_dpp(0, __builtin_bit_cast(int, x), 0x4E, 0xF, 0xF, true);
    x += __builtin_bit_cast(float, t);
    // row_half_mirror : pairs the two quads within each group of 8
    t = __builtin_amdgcn_update_dpp(0, __builtin_bit_cast(int, x), 0x141, 0xF, 0xF, true);
    x += __builtin_bit_cast(float, t);
    // row_mirror : pairs the two groups of 8 within the row of 16
    t = __builtin_amdgcn_update_dpp(0, __builtin_bit_cast(int, x), 0x140, 0xF, 0xF, true);
    x += __builtin_bit_cast(float, t);
    return x;
}

// One wave processes chunks of 16 points.
//   A (16x4 f32): combined weights M_b^T  (row = hidden unit k, K = homogeneous j)
//   B (4x16 f32): points^T                (col = point, K = homogeneous j)
//   D = A x B : lane (n16,hi), VGPR r  ->  h[point = n16][k = r + 8*hi]
// ReLU+dot is (h+|h|)*(W2/2) in-lane; cross-half add via permlanex16;
// scatter uses a wave-uniform single-atomic-per-pose fast path for runs.
__global__ __launch_bounds__(256) void sphere_proj_wmma_kernel(
    const float* __restrict__ T,     // [4,4,4] pose matrices
    const float* __restrict__ grid,  // [N,4] homogeneous points
    const float* __restrict__ W1,    // [3,16]
    const float* __restrict__ b1,    // [16]
    const float* __restrict__ W2,    // [16,1] -> flat [16]
    const float* __restrict__ b2,    // [1]
    const int*   __restrict__ seg,   // [N] sorted pixel id per point
    float* __restrict__ out,         // [4, 192*192]
    int N)
{
    const int lane = (int)(threadIdx.x & 31);
    const int n16  = lane & 15;   // point column (B/D), hidden row (A)
    const int hi   = lane >> 4;   // wave half: selects K pair {2hi, 2hi+1}

    // ---- Build per-pose A operand (combined 16x4 weight matrix, row = k) ----
    // M_b[j][k] = sum_{i<3} T[b][i][j] * W1[i][k]            (j = 0..2)
    // M_b[3][k] = sum_{i<3} T[b][i][3] * W1[i][k] + b1[k]    (translation + b1)
    // A layout: lane n16 = row k; VGPR0 = K=2hi, VGPR1 = K=2hi+1.
    const float w1a  = W1[0 * HIDDEN + n16];
    const float w1b  = W1[1 * HIDDEN + n16];
    const float w1c  = W1[2 * HIDDEN + n16];
    const float bias = b1[n16];

    v2f Aw[NPOSE];
#pragma unroll
    for (int b = 0; b < NPOSE; ++b) {
        const float* Tb = T + b * 16;
        const int j0 = 2 * hi;
        const int j1 = 2 * hi + 1;
        float ax = Tb[0 * 4 + j0] * w1a + Tb[1 * 4 + j0] * w1b + Tb[2 * 4 + j0] * w1c;
        float ay = Tb[0 * 4 + j1] * w1a + Tb[1 * 4 + j1] * w1b + Tb[2 * 4 + j1] * w1c;
        if (j1 == 3) ay += bias;   // K=3 row carries translation + b1
        Aw[b].x = ax;
        Aw[b].y = ay;
    }

    // Half of the second-layer weights for the 8 hidden units this half owns:
    // relu(h)*w == (h + |h|) * (w/2).
    float w2h[8];
#pragma unroll
    for (int r = 0; r < 8; ++r) w2h[r] = 0.5f * W2[r + 8 * hi];
    const float b2v = b2[0];

    const v2f* gp = (const v2f*)grid;   // [N,4] viewed as [N*2] float2

    const int nChunks = (N + 15) >> 4;
    const int wave    = (int)((blockIdx.x * blockDim.x + threadIdx.x) >> 5);
    const int nWaves  = (int)((gridDim.x * blockDim.x) >> 5);

    for (int c = wave; c < nChunks; c += nWaves) {
        const int base = c << 4;
        int idx = base + n16;
        if (idx >= N) idx = N - 1;           // clamp tail; guarded below

        // B operand: column n16 = point, VGPR0/1 = K = {2hi, 2hi+1}
        v2f pb = gp[idx * 2 + hi];
        // Pixel id for this lane's point (same pattern in both halves)
        int segv = seg[idx];

        // Wave-uniform run detection: segv is identical across the two halves,
        // so the ballot result (and this branch) is uniform across the wave.
        const int seg0 = __builtin_amdgcn_readfirstlane(segv);
        const bool fullChunk = (base + 16 <= N);
        const unsigned long long bal = __ballot(segv == seg0);
        const bool uniformRun = fullChunk && ((unsigned int)bal == 0xFFFFFFFFu);

        // ---- Compute all pose results first (single scatter region) ----
        float tv[NPOSE];
#pragma unroll
        for (int b = 0; b < NPOSE; ++b) {
            v8f d = {};
            // (neg_a, A, neg_b, B, c_mod, C, reuse_a, reuse_b)
            d = __builtin_amdgcn_wmma_f32_16x16x4_f32(
                    false, Aw[b], false, pb, (short)0, d, false, false);

            // s = sum_r relu(h[n16][r+8hi]) * W2[r+8hi]
            //   = sum_r (d[r] + |d[r]|) * (W2[r+8hi]/2)
            float s = 0.0f;
#pragma unroll
            for (int q = 0; q < 8; ++q) {
                s = fmaf(add_abs(d[q]), w2h[q], s);
            }
            // Combine halves: full 16-unit MLP output for point n16
            tv[b] = xor16_add(s);
        }

        // ---- Scatter ----
        if (uniformRun) {
            // All 16 points hit the same pixel: one atomic per pose.
            float tot[NPOSE];
#pragma unroll
            for (int b = 0; b < NPOSE; ++b) tot[b] = row16_reduce_add(tv[b]);
            if (lane == 0) {
#pragma unroll
                for (int b = 0; b < NPOSE; ++b) {
                    atomic_add_f32(out + b * HWPIX + seg0, tot[b] + 16.0f * b2v);
                }
            }
        } else {
            if (hi == 0 && (base + n16) < N) {
#pragma unroll
                for (int b = 0; b < NPOSE; ++b) {
                    atomic_add_f32(out + b * HWPIX + segv, tv[b] + b2v);
                }
            }
        }
    }
}

extern "C" void kernel_launch(void* const* d_in, const int* in_sizes, int n_in,
                              void* d_out, int out_size, void* d_ws, size_t ws_size,
                              hipStream_t stream) {
    // setup_inputs() order:
    //   0: transform_matrix [4,4,4] f32
    //   1: grid             [N,4]   f32
    //   2: W1               [3,16]  f32
    //   3: b1               [16]    f32
    //   4: W2               [16,1]  f32
    //   5: b2               [1]     f32
    //   6: seg_ids          [N]     i32
    //   7: num_pixels       scalar
    const float* T    = (const float*)d_in[0];
    const float* grid = (const float*)d_in[1];
    const float* W1   = (const float*)d_in[2];
    const float* b1   = (const float*)d_in[3];
    const float* W2   = (const float*)d_in[4];
    const float* b2   = (const float*)d_in[5];
    const int*   seg  = (const int*)d_in[6];
    float* out = (float*)d_out;

    const int N = in_sizes[1] / 4;

    // Output is accumulated with atomics; zero it first (graph-capturable).
    hipMemsetAsync(d_out, 0, (size_t)out_size * sizeof(float), stream);

    const int threads = 256;   // 8 waves per block
    const int blocks  = 2048;  // ~16K waves grid-striding over ~231K chunks
    sphere_proj_wmma_kernel<<<blocks, threads, 0, stream>>>(
        T, grid, W1, b1, W2, b2, seg, out, N);
}